// NanoMolmo2_19481971654845
// MI455X (gfx1250) — compile-verified
//
#include <hip/hip_runtime.h>
#include <hip/hip_bf16.h>
#include <stdint.h>

// Problem constants (match reference)
#define IMAGE_TOKEN_ID 31999
#define B_       4
#define L_       1024
#define D_       2048
#define NUM_VIS  576
#define NEW_LEN  (L_ - 1 + NUM_VIS)   // 1599

// ---- CDNA5 async global<->LDS DMA path (gfx1250) ------------------------
#if __has_builtin(__builtin_amdgcn_global_load_async_to_lds_b128) && \
    __has_builtin(__builtin_amdgcn_global_store_async_from_lds_b128)
#define HAVE_ASYNC_LDS 1
#else
#define HAVE_ASYNC_LDS 0
#endif

// gfx12/gfx1250 CPol: TH in bits [2:0]; TH=1 -> non-temporal.
#define CPOL_RT 0
#define CPOL_NT 1

// The builtins take pointers to a GCC-vector int4 in explicit address spaces:
//   param0: __attribute__((address_space(1))) v4i*   (global)
//   param1: __attribute__((address_space(3))) v4i*   (LDS)
typedef int v4i_ __attribute__((vector_size(16)));
typedef __attribute__((address_space(1))) v4i_* gv4p_;
typedef __attribute__((address_space(3))) v4i_* lv4p_;

// Generic -> AS casts via integer (generic shared-pointer low 32 bits are the
// hardware LDS byte address per ISA).
#define GAS4_(p) ((gv4p_)(uintptr_t)(p))
#define LAS4_(p) ((lv4p_)(unsigned)(uintptr_t)(p))

__device__ __forceinline__ void wait_asynccnt0() {
#if __has_builtin(__builtin_amdgcn_s_wait_asynccnt)
    __builtin_amdgcn_s_wait_asynccnt(0);
#else
    asm volatile("s_wait_asynccnt 0x0" ::: "memory");
#endif
}

// Main kernel (defined first so the disasm snippet shows it):
// one 256-thread block copies one 8KB output row (D_ floats).
// Row classification:
//   j <  ip            : text, input position j
//   ip <= j < ip+NV    : visual token (j - ip)
//   j >= ip+NV         : text, input position j - NUM_VIS + 1
__global__ void nm2_merge_copy(const int*   __restrict__ input_ids,
                               const float* __restrict__ visual,     // [B, NUM_VIS, D]
                               const float* __restrict__ embed,      // [32768, D]
                               const int*   __restrict__ img_pos,    // [B]
                               float*       __restrict__ merged,     // [B, NEW_LEN, D]
                               float*       __restrict__ pos_out) {  // [B, NEW_LEN]
    const int row = blockIdx.x;
    const int b   = row / NEW_LEN;
    const int j   = row - b * NEW_LEN;
    const int ip  = img_pos[b];

    const float* src;
    if (j < ip) {
        src = embed + (size_t)input_ids[b * L_ + j] * D_;
    } else if (j < ip + NUM_VIS) {
        src = visual + ((size_t)b * NUM_VIS + (size_t)(j - ip)) * D_;
    } else {
        src = embed + (size_t)input_ids[b * L_ + (j - NUM_VIS + 1)] * D_;
    }
    float* dst = merged + (size_t)row * D_;

    if (threadIdx.x == 0) {
        pos_out[row] = (float)j;   // position_ids value (flat float out-buffer)
    }

    const int t = threadIdx.x;

#if HAVE_ASYNC_LDS
    // Stage the row through LDS with the async DMA path (ASYNCcnt-tracked,
    // no VGPR round trip). 512 x 16B chunks; thread t owns chunks t, t+256.
    __shared__ __align__(16) float lds[D_];
    const char* s = (const char*)src;
    char*       l = (char*)lds;
    char*       d = (char*)dst;

    __builtin_amdgcn_global_load_async_to_lds_b128(GAS4_(s + (size_t)t * 16),
                                                   LAS4_(l + (size_t)t * 16), 0, CPOL_RT);
    __builtin_amdgcn_global_load_async_to_lds_b128(GAS4_(s + (size_t)(t + 256) * 16),
                                                   LAS4_(l + (size_t)(t + 256) * 16), 0, CPOL_RT);
    // Each thread stores only chunks it loaded -> per-wave wait suffices,
    // no workgroup barrier needed.
    wait_asynccnt0();
    // Output is write-once/streamed: non-temporal stores, don't churn L2.
    __builtin_amdgcn_global_store_async_from_lds_b128(GAS4_(d + (size_t)t * 16),
                                                      LAS4_(l + (size_t)t * 16), 0, CPOL_NT);
    __builtin_amdgcn_global_store_async_from_lds_b128(GAS4_(d + (size_t)(t + 256) * 16),
                                                      LAS4_(l + (size_t)(t + 256) * 16), 0, CPOL_NT);
    // Outstanding ASYNCcnt drained by S_ENDPGM's implicit S_WAIT_IDLE.
#else
    // Fallback: plain B128 register copies.
    const float4* s4 = (const float4*)src;
    float4*       d4 = (float4*)dst;
    d4[t]       = s4[t];
    d4[t + 256] = s4[t + 256];
#endif
}

// Pass 1: locate the single image token per batch row (tiny, runs once over
// 16KB of ids; keep the loop rolled to avoid pointless code bloat).
__global__ void nm2_find_img_pos(const int* __restrict__ input_ids,
                                 int* __restrict__ img_pos) {
    const int b = blockIdx.x;
#pragma clang loop unroll(disable)
    for (int i = threadIdx.x; i < L_; i += 256) {
        if (input_ids[b * L_ + i] == IMAGE_TOKEN_ID) img_pos[b] = i;  // exactly one hit
    }
}

extern "C" void kernel_launch(void* const* d_in, const int* in_sizes, int n_in,
                              void* d_out, int out_size, void* d_ws, size_t ws_size,
                              hipStream_t stream) {
    // setup_inputs() order: input_ids (int32), visual_tokens (f32), embed_table (f32)
    const int*   input_ids = (const int*)d_in[0];
    const float* visual    = (const float*)d_in[1];
    const float* embed     = (const float*)d_in[2];

    float* merged  = (float*)d_out;                          // [B, NEW_LEN, D]
    float* pos_out = merged + (size_t)B_ * NEW_LEN * D_;     // [B, NEW_LEN]
    int*   img_pos = (int*)d_ws;                             // [B]

    nm2_find_img_pos<<<B_, 256, 0, stream>>>(input_ids, img_pos);
    nm2_merge_copy<<<B_ * NEW_LEN, 256, 0, stream>>>(input_ids, visual, embed,
                                                     img_pos, merged, pos_out);
}